// UnGCNLayer_12953621365354
// MI455X (gfx1250) — compile-verified
//
#include <hip/hip_runtime.h>
#include <hip/hip_bf16.h>
#include <math.h>

typedef __attribute__((ext_vector_type(2))) float v2f;
typedef __attribute__((ext_vector_type(8))) float v8f;

#define HDIM 64
#define TILE_M 16

// ---------------------------------------------------------------------------
// Kernel 1: init output accumulators with the bias vectors (segment_sum + b).
// ---------------------------------------------------------------------------
__global__ void ugcn_init_kernel(float* __restrict__ Om, float* __restrict__ Os,
                                 const float* __restrict__ bm,
                                 const float* __restrict__ bs, int n) {
  int i = blockIdx.x * blockDim.x + threadIdx.x;
  if (i < n) {
    int f = i & (HDIM - 1);
    Om[i] = bm[f];
    Os[i] = bs[f];
  }
}

// ---------------------------------------------------------------------------
// Kernel 2: h = x @ W via V_WMMA_F32_16X16X4_F32.
// blockIdx.y == 0 : A = mean,     W = Wm_mu -> Hm
// blockIdx.y == 1 : A = std^2,    W = Ws_mu -> Hs  (var = std*std fused here)
// One wave computes one 16-row tile across all 64 output columns.
// ---------------------------------------------------------------------------
__global__ void ugcn_gemm_wmma_kernel(const float* __restrict__ A0,
                                      const float* __restrict__ A1,
                                      const float* __restrict__ W0,
                                      const float* __restrict__ W1,
                                      float* __restrict__ H0,
                                      float* __restrict__ H1,
                                      int ntiles) {
  const int wave = threadIdx.x >> 5;
  const int lane = threadIdx.x & 31;
  const int tile = blockIdx.x * (blockDim.x >> 5) + wave;
  if (tile >= ntiles) return;  // uniform across the wave -> EXEC stays all-1s

  const bool sq = (blockIdx.y != 0);
  const float* __restrict__ A = sq ? A1 : A0;
  const float* __restrict__ W = sq ? W1 : W0;
  float* __restrict__ H       = sq ? H1 : H0;

  const int mbase = tile * TILE_M;
  const int mlane = lane & 15;            // M (A) or N (B/C) within the tile
  const int khalf = (lane >> 4) * 2;      // lanes 0-15 -> K{0,1}; 16-31 -> K{2,3}

  // Preload all 16 A fragments (16x4 f32 tiles covering K=0..63).
  // ISA 7.12.2: A 16x4 f32 -> 2 VGPRs; lane L<16 holds {K0,K1}, L>=16 {K2,K3}.
  v2f afrag[16];
  const float* __restrict__ arow = A + (size_t)(mbase + mlane) * HDIM;
#pragma unroll
  for (int kb = 0; kb < 16; ++kb) {
    float ax = arow[kb * 4 + khalf + 0];
    float ay = arow[kb * 4 + khalf + 1];
    if (sq) { ax *= ax; ay *= ay; }       // variance path: square on load
    afrag[kb].x = ax;
    afrag[kb].y = ay;
  }

#pragma unroll
  for (int nt = 0; nt < 4; ++nt) {        // four 16-column output tiles
    const int ncol = nt * 16 + mlane;
    v8f c = {};
#pragma unroll
    for (int kb = 0; kb < 16; ++kb) {
      // B 4x16 f32 -> 2 VGPRs; lane L<16 row-pair {K0,K1}, L>=16 {K2,K3}.
      v2f b;
      b.x = W[(size_t)(kb * 4 + khalf + 0) * HDIM + ncol];
      b.y = W[(size_t)(kb * 4 + khalf + 1) * HDIM + ncol];
      c = __builtin_amdgcn_wmma_f32_16x16x4_f32(
          /*neg_a=*/false, afrag[kb], /*neg_b=*/false, b,
          /*c_mod=*/(short)0, c, /*reuse_a=*/false, /*reuse_b=*/false);
    }
    // C/D 16x16 f32 layout: lane<16 -> N=lane, VGPR v holds M=v;
    // lane>=16 -> N=lane-16, VGPR v holds M=v+8.
    const int m0 = (lane >> 4) * 8;
#pragma unroll
    for (int v = 0; v < 8; ++v) {
      H[(size_t)(mbase + m0 + v) * HDIM + ncol] = c[v];
    }
  }
}

// ---------------------------------------------------------------------------
// Kernel 3: edge scatter. 16 threads per edge, float4 per thread.
//   out_mean[dst]   += h_m[src] * w
//   out_logvar[dst] += h_s[src] * w*w
// Edge-list loads are non-temporal (streamed once); h/out stay L2-resident.
// ---------------------------------------------------------------------------
__global__ void ugcn_scatter_kernel(const int* __restrict__ ei,
                                    const float* __restrict__ ew,
                                    const float* __restrict__ Hm,
                                    const float* __restrict__ Hs,
                                    float* __restrict__ Om,
                                    float* __restrict__ Os, int E) {
  long long t = (long long)blockIdx.x * blockDim.x + threadIdx.x;
  int e = (int)(t >> 4);
  if (e >= E) return;
  int f4 = (int)(t & 15) * 4;

  int src = __builtin_nontemporal_load(ei + e);
  int dst = __builtin_nontemporal_load(ei + E + e);
  float w = __builtin_nontemporal_load(ew + e);
  float w2 = w * w;

  const float4 hm = *(const float4*)(Hm + (size_t)src * HDIM + f4);
  const float4 hs = *(const float4*)(Hs + (size_t)src * HDIM + f4);
  float* om = Om + (size_t)dst * HDIM + f4;
  float* os = Os + (size_t)dst * HDIM + f4;

  atomicAdd(om + 0, hm.x * w);
  atomicAdd(om + 1, hm.y * w);
  atomicAdd(om + 2, hm.z * w);
  atomicAdd(om + 3, hm.w * w);
  atomicAdd(os + 0, hs.x * w2);
  atomicAdd(os + 1, hs.y * w2);
  atomicAdd(os + 2, hs.z * w2);
  atomicAdd(os + 3, hs.w * w2);
}

// ---------------------------------------------------------------------------
// Kernel 4: new_std = sqrt(exp(new_log_var) + 1e-6), in place.
// ---------------------------------------------------------------------------
__global__ void ugcn_finalize_kernel(float* __restrict__ Os, int n) {
  int i = blockIdx.x * blockDim.x + threadIdx.x;
  if (i < n) Os[i] = sqrtf(__expf(Os[i]) + 1e-6f);
}

// ---------------------------------------------------------------------------
// Kernel 5: total KL for both weight matrices (single block, LDS reduction).
// KL elem = log(0.1) - ls + (exp(2*ls) + mu^2) * 50 - 0.5
// ---------------------------------------------------------------------------
__global__ void ugcn_kl_kernel(const float* __restrict__ Wm_mu,
                               const float* __restrict__ Wm_ls,
                               const float* __restrict__ Ws_mu,
                               const float* __restrict__ Ws_ls,
                               float* __restrict__ out) {
  __shared__ float red[256];
  const float logp = -2.3025850929940457f;  // log(0.1)
  float s = 0.0f;
  for (int i = threadIdx.x; i < HDIM * HDIM; i += 256) {
    {
      float mu = Wm_mu[i], ls = Wm_ls[i];
      float sg = __expf(ls);
      s += logp - ls + (sg * sg + mu * mu) * 50.0f - 0.5f;
    }
    {
      float mu = Ws_mu[i], ls = Ws_ls[i];
      float sg = __expf(ls);
      s += logp - ls + (sg * sg + mu * mu) * 50.0f - 0.5f;
    }
  }
  red[threadIdx.x] = s;
  __syncthreads();
  for (int off = 128; off > 0; off >>= 1) {
    if ((int)threadIdx.x < off) red[threadIdx.x] += red[threadIdx.x + off];
    __syncthreads();
  }
  if (threadIdx.x == 0) out[0] = red[0];
}

// ---------------------------------------------------------------------------
extern "C" void kernel_launch(void* const* d_in, const int* in_sizes, int n_in,
                              void* d_out, int out_size, void* d_ws, size_t ws_size,
                              hipStream_t stream) {
  const float* mean  = (const float*)d_in[0];
  const float* stdv  = (const float*)d_in[1];
  const int*   ei    = (const int*)d_in[2];
  const float* ew    = (const float*)d_in[3];
  const float* Wm_mu = (const float*)d_in[4];
  const float* Wm_ls = (const float*)d_in[5];
  const float* bm    = (const float*)d_in[6];
  const float* Ws_mu = (const float*)d_in[7];
  const float* Ws_ls = (const float*)d_in[8];
  const float* bs    = (const float*)d_in[9];

  const int N  = in_sizes[0] / HDIM;   // 50000
  const int NH = N * HDIM;             // 3,200,000
  const int E  = in_sizes[3];          // 1,600,000

  float* out = (float*)d_out;
  float* Om = out;            // new_mean accumulator / final output
  float* Os = out + NH;       // new_log_var accumulator -> new_std in place
  float* kl = out + 2 * NH;   // scalar total_kl

  float* Hm = (float*)d_ws;   // mean @ Wm_mu   [N, 64]
  float* Hs = Hm + NH;        // var  @ Ws_mu   [N, 64]

  // 1) bias init of accumulators
  ugcn_init_kernel<<<(NH + 255) / 256, 256, 0, stream>>>(Om, Os, bm, bs, NH);

  // 2) both GEMMs via f32 WMMA (blockIdx.y selects mean / variance path)
  const int ntiles = N / TILE_M;                 // 3125
  dim3 ggrid((ntiles + 7) / 8, 2);
  ugcn_gemm_wmma_kernel<<<ggrid, 256, 0, stream>>>(mean, stdv, Wm_mu, Ws_mu,
                                                   Hm, Hs, ntiles);

  // 3) edge scatter-add (L2-resident atomics)
  long long sthreads = (long long)E * 16;
  int sblocks = (int)((sthreads + 255) / 256);
  ugcn_scatter_kernel<<<sblocks, 256, 0, stream>>>(ei, ew, Hm, Hs, Om, Os, E);

  // 4) new_std = sqrt(exp(log_var) + 1e-6)
  ugcn_finalize_kernel<<<(NH + 255) / 256, 256, 0, stream>>>(Os, NH);

  // 5) total KL scalar
  ugcn_kl_kernel<<<1, 256, 0, stream>>>(Wm_mu, Wm_ls, Ws_mu, Ws_ls, kl);
}